// Scatter_48172353192534
// MI455X (gfx1250) — compile-verified
//
#include <hip/hip_runtime.h>

#ifndef __has_builtin
#define __has_builtin(x) 0
#endif

#define NN 50000      // nodes
#define NE 800000     // edges
#define CI 32         // input channels
#define C2 128        // channels in second diffusion (C*J)
#define NG 64         // graphs
#define NF 352        // 11*32 features per node

typedef __attribute__((ext_vector_type(2))) float v2f;
typedef __attribute__((ext_vector_type(8))) float v8f;

// ---------------- elementwise / setup kernels (float4 data path) ----------

__global__ void k_zero4(float4* __restrict__ p, int n4) {
  const float4 z = {0.f, 0.f, 0.f, 0.f};
  for (int i = blockIdx.x * blockDim.x + threadIdx.x; i < n4;
       i += gridDim.x * blockDim.x)
    p[i] = z;
}

__global__ void k_deg(const int* __restrict__ col, float* __restrict__ deg) {
  int e = blockIdx.x * blockDim.x + threadIdx.x;
  if (e < NE) atomicAdd(&deg[col[e]], 1.0f);
}

__global__ void k_recip(float* __restrict__ d) {
  int i = blockIdx.x * blockDim.x + threadIdx.x;
  if (i < NN) {
    float v = d[i];
    d[i] = (v > 0.0f) ? (1.0f / v) : 0.0f;
  }
}

// carry0 = x ; g row0 = x   (i over NN*8 float4 chunks)
__global__ void k_init_x(const float4* __restrict__ x, float4* __restrict__ carry,
                         float4* __restrict__ g4) {
  int i = blockIdx.x * blockDim.x + threadIdx.x;
  if (i < NN * (CI / 4)) {
    int n = i >> 3, c4 = i & 7;
    float4 v = x[i];
    carry[i] = v;
    g4[n * (NF / 4) + c4] = v;
  }
}

// agg[col] += dinv[col] * carry[row], float4 payload per thread, f32 atomics
__global__ void k_scatter(const float4* __restrict__ src, float* __restrict__ dst,
                          const int* __restrict__ row, const int* __restrict__ col,
                          const float* __restrict__ dinv, int cvec, int C) {
  long long total = (long long)NE * cvec;
  long long stride = (long long)gridDim.x * blockDim.x;
  for (long long i = blockIdx.x * (long long)blockDim.x + threadIdx.x; i < total;
       i += stride) {
    int e = (int)(i / cvec);
    int q = (int)(i % cvec);
    int r = row[e], cl = col[e];
    float w = dinv[cl];
    float4 v = src[(long long)r * cvec + q];
    float* d = dst + (long long)cl * C + q * 4;
    atomicAdd(d + 0, w * v.x);
    atomicAdd(d + 1, w * v.y);
    atomicAdd(d + 2, w * v.z);
    atomicAdd(d + 3, w * v.w);
  }
}

// nxt = 0.5*(carry + agg), written into agg (becomes new carry)
__global__ void k_combine4(const float4* __restrict__ oldc, float4* __restrict__ agg,
                           int n4) {
  for (int i = blockIdx.x * blockDim.x + threadIdx.x; i < n4;
       i += gridDim.x * blockDim.x) {
    float4 a = oldc[i], b = agg[i];
    float4 r = {0.5f * (a.x + b.x), 0.5f * (a.y + b.y),
                0.5f * (a.z + b.z), 0.5f * (a.w + b.w)};
    agg[i] = r;
  }
}

// d1 snapshot: jslot<0 -> only save prev; else g row (1+jslot) = |cur - prev|
__global__ void k_snap32(const float4* __restrict__ cur, float4* __restrict__ prev,
                         float4* __restrict__ g4, int jslot) {
  int i = blockIdx.x * blockDim.x + threadIdx.x;
  if (i < NN * (CI / 4)) {
    int n = i >> 3, c4 = i & 7;
    float4 v = cur[i];
    if (jslot >= 0) {
      float4 p = prev[i];
      float4 d = {fabsf(v.x - p.x), fabsf(v.y - p.y),
                  fabsf(v.z - p.z), fabsf(v.w - p.w)};
      g4[n * (NF / 4) + (1 + jslot) * (CI / 4) + c4] = d;
    }
    prev[i] = v;
  }
}

// d2 snapshot at step 2^(w+1): float4 = (j=0..3) for one channel c.
// feng-filtered |sc2| entries -> g rows 5..10:
//   w=1: j0->row5 ; w=2: j0->row6, j1->row7 ; w=3: j0->row8, j1->row9, j2->row10
__global__ void k_snap128(const float4* __restrict__ cur, float4* __restrict__ prev,
                          float* __restrict__ g, int w) {
  int i = blockIdx.x * blockDim.x + threadIdx.x;
  if (i < NN * (C2 / 4)) {
    int n = i >> 5, c = i & 31;
    float4 v = cur[i];
    float4 p = prev[i];
    float* gr = g + (long long)n * NF + c;
    if (w == 1) {
      gr[5 * CI] = fabsf(v.x - p.x);
    } else if (w == 2) {
      gr[6 * CI] = fabsf(v.x - p.x);
      gr[7 * CI] = fabsf(v.y - p.y);
    } else if (w == 3) {
      gr[8 * CI]  = fabsf(v.x - p.x);
      gr[9 * CI]  = fabsf(v.y - p.y);
      gr[10 * CI] = fabsf(v.z - p.z);
    }
    prev[i] = v;
  }
}

// build d2 initial carry (layout [n, c*4+j]) from S1 stored in g rows 1..4
__global__ void k_buildS1(const float* __restrict__ g, float4* __restrict__ carry) {
  int i = blockIdx.x * blockDim.x + threadIdx.x;
  if (i < NN * (C2 / 4)) {
    int n = i >> 5, c = i & 31;
    const float* gr = g + (long long)n * NF + c;
    float4 v = {gr[1 * CI], gr[2 * CI], gr[3 * CI], gr[4 * CI]};
    carry[i] = v;
  }
}

// ---------------- per-graph moments ----------------

__global__ void k_binit(int* starts, int* ends, float* cnt) {
  int i = threadIdx.x;
  if (i < NG) { starts[i] = NN; ends[i] = 0; cnt[i] = 0.0f; }
}

__global__ void k_bounds(const int* __restrict__ batch, int* starts, int* ends,
                         float* cnt) {
  int i = blockIdx.x * blockDim.x + threadIdx.x;
  if (i < NN) {
    int gb = batch[i];
    atomicMin(&starts[gb], i);
    atomicMax(&ends[gb], i + 1);
    atomicAdd(&cnt[gb], 1.0f);
  }
}

// One wave per (graph, 16-feature chunk): power sums S1..S4 via WMMA.
// D(16x16) += ones(16x4) * B(4 nodes x 16 feats): every row of D holds the
// per-feature column sums. 4 accumulators = powers x, x^2, x^3, x^4.
// Main loop is bounds-check free (uniform trip count); single guarded tail.
__global__ void __launch_bounds__(32)
k_moments(const float* __restrict__ g, const int* __restrict__ starts,
          const int* __restrict__ ends, float* __restrict__ sums) {
  int gr = blockIdx.y;
  int fb = blockIdx.x * 16;
  int lane = threadIdx.x & 31;
  int f = fb + (lane & 15);
  int koff = (lane < 16) ? 0 : 2;  // B rows: lanes 0-15 hold K=0,1; 16-31 K=2,3
  int s = starts[gr], e = ends[gr];
  int len = e - s;
  if (len < 0) len = 0;
  int main_end = s + (len & ~3);

#if __has_builtin(__builtin_amdgcn_wmma_f32_16x16x4_f32)
  v8f a1 = {0,0,0,0,0,0,0,0};
  v8f a2 = {0,0,0,0,0,0,0,0};
  v8f a3 = {0,0,0,0,0,0,0,0};
  v8f a4 = {0,0,0,0,0,0,0,0};
  v2f one = {1.0f, 1.0f};
  int base = s;
  for (; base < main_end; base += 4) {
    const float* p0 = g + (long long)(base + koff) * NF + f;
    float x0 = p0[0];
    float x1 = p0[NF];
    __builtin_prefetch(p0 + 4 * NF, 0, 1);   // next chunk (global_prefetch_b8)
    v2f b = {x0, x1};
    v2f b2 = b * b;
    v2f b3 = b2 * b;
    v2f b4 = b2 * b2;
    a1 = __builtin_amdgcn_wmma_f32_16x16x4_f32(false, one, false, b,  (short)0, a1, false, false);
    a2 = __builtin_amdgcn_wmma_f32_16x16x4_f32(false, one, false, b2, (short)0, a2, false, false);
    a3 = __builtin_amdgcn_wmma_f32_16x16x4_f32(false, one, false, b3, (short)0, a3, false, false);
    a4 = __builtin_amdgcn_wmma_f32_16x16x4_f32(false, one, false, b4, (short)0, a4, false, false);
  }
  if (base < e) {  // uniform tail (<=1 iteration)
    int n0 = base + koff, n1 = n0 + 1;
    float x0 = (n0 < e) ? g[(long long)n0 * NF + f] : 0.0f;
    float x1 = (n1 < e) ? g[(long long)n1 * NF + f] : 0.0f;
    v2f b = {x0, x1};
    v2f b2 = b * b;
    v2f b3 = b2 * b;
    v2f b4 = b2 * b2;
    a1 = __builtin_amdgcn_wmma_f32_16x16x4_f32(false, one, false, b,  (short)0, a1, false, false);
    a2 = __builtin_amdgcn_wmma_f32_16x16x4_f32(false, one, false, b2, (short)0, a2, false, false);
    a3 = __builtin_amdgcn_wmma_f32_16x16x4_f32(false, one, false, b3, (short)0, a3, false, false);
    a4 = __builtin_amdgcn_wmma_f32_16x16x4_f32(false, one, false, b4, (short)0, a4, false, false);
  }
  if (lane < 16) {
    long long o = (long long)gr * 4 * NF + f;
    sums[o]          = a1[0];   // D row M=0, col N=lane
    sums[o + NF]     = a2[0];
    sums[o + 2 * NF] = a3[0];
    sums[o + 3 * NF] = a4[0];
  }
#else
  float s1 = 0.f, s2 = 0.f, s3 = 0.f, s4 = 0.f;
  for (int b0 = s; b0 < e; b0 += 4) {
    int n0 = b0 + koff, n1 = n0 + 1;
    float x0 = (n0 < e) ? g[(long long)n0 * NF + f] : 0.0f;
    float x1 = (n1 < e) ? g[(long long)n1 * NF + f] : 0.0f;
    float p0 = x0 * x0, p1 = x1 * x1;
    s1 += x0 + x1;
    s2 += p0 + p1;
    s3 += p0 * x0 + p1 * x1;
    s4 += p0 * p0 + p1 * p1;
  }
  s1 += __shfl_xor(s1, 16, 32);
  s2 += __shfl_xor(s2, 16, 32);
  s3 += __shfl_xor(s3, 16, 32);
  s4 += __shfl_xor(s4, 16, 32);
  if (lane < 16) {
    long long o = (long long)gr * 4 * NF + f;
    sums[o] = s1; sums[o + NF] = s2; sums[o + 2 * NF] = s3; sums[o + 3 * NF] = s4;
  }
#endif
}

__global__ void k_final(const float* __restrict__ sums, const float* __restrict__ cnt,
                        float* __restrict__ out) {
  int i = blockIdx.x * blockDim.x + threadIdx.x;
  if (i < NG * NF) {
    int gr = i / NF, f = i % NF;
    float n = fmaxf(cnt[gr], 1.0f);
    float inv = 1.0f / n;
    const float* S = sums + (long long)gr * 4 * NF;
    float m  = S[f] * inv;
    float e2 = S[NF + f] * inv;
    float e3 = S[2 * NF + f] * inv;
    float e4 = S[3 * NF + f] * inv;
    float var = e2 - m * m;
    float m3 = e3 - 3.0f * m * e2 + 2.0f * m * m * m;
    float m4 = e4 - 4.0f * m * e3 + 6.0f * m * m * e2 - 3.0f * m * m * m * m;
    const float BIG = 1e15f;
    float sk = m3 / (var * sqrtf(var));
    if ((sk != sk) || sk > BIG) sk = 0.0f;
    float ku = m4 / (var * var) - 3.0f;
    if ((ku != ku) || ku > BIG) ku = -3.0f;
    float* o = out + (long long)gr * (4 * NF);
    o[f] = m;
    o[NF + f] = var;
    o[2 * NF + f] = sk;
    o[3 * NF + f] = ku;
  }
}

__global__ void k_wav(const float* __restrict__ w, float* __restrict__ out) {
  int i = threadIdx.x;
  if (i < 68) out[i] = w[i];
}

// ---------------- host launch ----------------

extern "C" void kernel_launch(void* const* d_in, const int* in_sizes, int n_in,
                              void* d_out, int out_size, void* d_ws, size_t ws_size,
                              hipStream_t stream) {
  (void)in_sizes; (void)n_in; (void)out_size; (void)ws_size;
  const float* x     = (const float*)d_in[0];
  const int*   ei    = (const int*)d_in[1];
  const int*   batch = (const int*)d_in[2];
  const float* wav   = (const float*)d_in[3];
  const int* row = ei;
  const int* col = ei + NE;

  float* ws = (float*)d_ws;
  size_t off = 0;
  auto take = [&](size_t nf) -> float* {
    float* p = ws + off;
    off = (off + nf + 63) & ~(size_t)63;   // 256B-aligned chunks
    return p;
  };
  float* dinv = take(NN);
  float* buf0 = take((size_t)NN * C2);
  float* buf1 = take((size_t)NN * C2);
  float* prev = take((size_t)NN * C2);
  float* g    = take((size_t)NN * NF);
  float* sums = take((size_t)NG * 4 * NF);
  float* cnt  = take(NG);
  int* starts = (int*)take(NG);
  int* ends   = (int*)take(NG);

  const int B = 256;
  const int n4_32  = NN * (CI / 4);   // 400000
  const int n4_128 = NN * (C2 / 4);   // 1600000

  // degrees -> dinv
  k_zero4<<<128, B, 0, stream>>>((float4*)dinv, NN / 4);
  k_deg<<<(NE + B - 1) / B, B, 0, stream>>>(col, dinv);
  k_recip<<<(NN + B - 1) / B, B, 0, stream>>>(dinv);

  // d1 diffusion (C=32), snapshots at 1,2,4,8,16
  k_init_x<<<(n4_32 + B - 1) / B, B, 0, stream>>>((const float4*)x,
                                                  (float4*)buf0, (float4*)g);
  float* cur = buf0;
  float* oth = buf1;
  for (int s = 1; s <= 16; ++s) {
    k_zero4<<<1024, B, 0, stream>>>((float4*)oth, n4_32);
    k_scatter<<<8192, B, 0, stream>>>((const float4*)cur, oth, row, col, dinv,
                                      CI / 4, CI);
    k_combine4<<<1024, B, 0, stream>>>((const float4*)cur, (float4*)oth, n4_32);
    float* t = cur; cur = oth; oth = t;
    int jslot = (s == 1) ? -1 : (s == 2) ? 0 : (s == 4) ? 1
               : (s == 8) ? 2 : (s == 16) ? 3 : -2;
    if (jslot >= -1)
      k_snap32<<<(n4_32 + B - 1) / B, B, 0, stream>>>((const float4*)cur,
                                                      (float4*)prev, (float4*)g,
                                                      jslot);
  }

  // d2 diffusion (C=128 = C_IN * J), snapshots at 2,4,8,16
  k_buildS1<<<(n4_128 + B - 1) / B, B, 0, stream>>>(g, (float4*)buf0);
  cur = buf0; oth = buf1;
  for (int s = 1; s <= 16; ++s) {
    k_zero4<<<4096, B, 0, stream>>>((float4*)oth, n4_128);
    k_scatter<<<16384, B, 0, stream>>>((const float4*)cur, oth, row, col, dinv,
                                       C2 / 4, C2);
    k_combine4<<<4096, B, 0, stream>>>((const float4*)cur, (float4*)oth, n4_128);
    float* t = cur; cur = oth; oth = t;
    int w = (s == 2) ? 0 : (s == 4) ? 1 : (s == 8) ? 2 : (s == 16) ? 3 : -2;
    if (w >= 0)
      k_snap128<<<(n4_128 + B - 1) / B, B, 0, stream>>>((const float4*)cur,
                                                        (float4*)prev, g, w);
  }

  // per-graph moments (WMMA reduction) + output
  k_binit<<<1, 64, 0, stream>>>(starts, ends, cnt);
  k_bounds<<<(NN + B - 1) / B, B, 0, stream>>>(batch, starts, ends, cnt);
  k_moments<<<dim3(NF / 16, NG), 32, 0, stream>>>(g, starts, ends, sums);
  k_final<<<(NG * NF + B - 1) / B, B, 0, stream>>>(sums, cnt, (float*)d_out);
  k_wav<<<1, 128, 0, stream>>>(wav, (float*)d_out + NG * 4 * NF);
}